// PositionalEncoding_3521873183416
// MI455X (gfx1250) — compile-verified
//
#include <hip/hip_runtime.h>
#include <cstdint>
#include <cstddef>

// out[b,t,f] = x[b,t,f] + P[t,f]
//   P[2g,   i] = sin(g / n^(2i/F))
//   P[2g+1, i] = cos(g / n^(2i/F))
// B=32, T=2048, F=1024, n=10000

#define PE_T 2048
#define PE_F 1024
#define PE_B 32

typedef float v4f __attribute__((ext_vector_type(4)));

// 2*log2(10000)/1024 (double-evaluated constant, rounded to f32 at compile time)
#define PE_C 0.02595256324130751843f

// Compute the 4 P values owned by (row t, cols i0..i0+3).  Force-inlined so
// kernels don't pay an out-of-line call (previous round emitted _Z11pe_compute4ii
// as a real function with frame setup).
static __device__ __attribute__((always_inline)) inline v4f pe_compute4(int t, int i0) {
    const float g = (float)(t >> 1);
    const bool is_cos = (t & 1) != 0;
    float v[4];
    #pragma unroll
    for (int k = 0; k < 4; ++k) {
        float i = (float)(i0 + k);
        float den = exp2f(PE_C * i);          // n^(2i/F) = 2^(i * 2*log2(n)/F)
        float ang = g / den;
        v[k] = is_cos ? cosf(ang) : sinf(ang);
    }
    v4f p;
    p.x = v[0]; p.y = v[1]; p.z = v[2]; p.w = v[3];
    return p;
}

// Kernel 1: precompute P[t,f] into workspace (8 MB, stays L2-resident).
__global__ void __launch_bounds__(256) pe_table_kernel(float* __restrict__ P) {
    const int t  = blockIdx.x;          // 0..2047
    const int i0 = threadIdx.x * 4;     // 0..1020
    v4f p = pe_compute4(t, i0);
    *(v4f*)(P + (size_t)t * PE_F + i0) = p;
}

// Kernel 2: each block owns one t-row and 8 batches. Stage the 4 KB P row
// into LDS with the CDNA5 async-to-LDS path, then stream x/out with NT hints.
__global__ void __launch_bounds__(256)
pe_add_async_kernel(const float* __restrict__ x,
                    const float* __restrict__ P,
                    float* __restrict__ out) {
    const int t   = blockIdx.x;         // row 0..2047
    const int bq  = blockIdx.y;         // batch quarter 0..3
    const int tid = threadIdx.x;        // 0..255

    __shared__ __align__(16) float sp[PE_F];   // 4 KB P row

    // Async copy: each of the 256 lanes moves 16 bytes (B128) -> 4096 B total.
    {
        const float* gsrc = P + (size_t)t * PE_F + tid * 4;
        uint32_t lds_off = (uint32_t)(uintptr_t)(&sp[0]) + (uint32_t)(tid * 16);
        asm volatile("global_load_async_to_lds_b128 %0, %1, off"
                     :: "v"(lds_off), "v"(gsrc) : "memory");
#if __has_builtin(__builtin_amdgcn_s_wait_asynccnt)
        __builtin_amdgcn_s_wait_asynccnt(0);
#else
        asm volatile("s_wait_asynccnt 0x0" ::: "memory");
#endif
    }
    __syncthreads();

    const v4f p = *(const v4f*)(sp + tid * 4);             // ds_load_b128
    const size_t rowbase = (size_t)t * PE_F + (size_t)(tid * 4);

    #pragma unroll
    for (int bb = 0; bb < PE_B / 4; ++bb) {                // 8 batches per block
        const int b = bq * (PE_B / 4) + bb;
        const size_t idx = (size_t)b * (size_t)PE_T * PE_F + rowbase;
        v4f xv = __builtin_nontemporal_load((const v4f*)(x + idx));   // th:NT, don't pollute L2
        v4f ov = xv + p;
        __builtin_nontemporal_store(ov, (v4f*)(out + idx));           // th:NT streaming store
    }
}

// Fallback (no workspace): fuse P computation in-registers, same streaming loop.
__global__ void __launch_bounds__(256)
pe_add_fused_kernel(const float* __restrict__ x, float* __restrict__ out) {
    const int t   = blockIdx.x;
    const int bq  = blockIdx.y;
    const int tid = threadIdx.x;
    const int i0  = tid * 4;

    const v4f p = pe_compute4(t, i0);
    const size_t rowbase = (size_t)t * PE_F + (size_t)i0;

    #pragma unroll
    for (int bb = 0; bb < PE_B / 4; ++bb) {
        const int b = bq * (PE_B / 4) + bb;
        const size_t idx = (size_t)b * (size_t)PE_T * PE_F + rowbase;
        v4f xv = __builtin_nontemporal_load((const v4f*)(x + idx));
        v4f ov = xv + p;
        __builtin_nontemporal_store(ov, (v4f*)(out + idx));
    }
}

extern "C" void kernel_launch(void* const* d_in, const int* in_sizes, int n_in,
                              void* d_out, int out_size, void* d_ws, size_t ws_size,
                              hipStream_t stream) {
    const float* x = (const float*)d_in[0];
    float* out = (float*)d_out;

    const size_t p_bytes = (size_t)PE_T * PE_F * sizeof(float);   // 8 MB

    if (ws_size >= p_bytes && d_ws != nullptr) {
        float* P = (float*)d_ws;
        pe_table_kernel<<<dim3(PE_T), dim3(256), 0, stream>>>(P);
        pe_add_async_kernel<<<dim3(PE_T, 4), dim3(256), 0, stream>>>(x, P, out);
    } else {
        pe_add_fused_kernel<<<dim3(PE_T, 4), dim3(256), 0, stream>>>(x, out);
    }
}